// BiRNN_28183575396934
// MI455X (gfx1250) — compile-verified
//
#include <hip/hip_runtime.h>
#include <hip/hip_bf16.h>

// ---------------------------------------------------------------------------
// BiRNN on gfx1250 (MI455X).
// Phase 1: bidirectional Elman RNN (5->9, T=2048) via V_WMMA_F32_16X16X4_F32.
//   Weights in A, data in B, hidden state in D.  u-slot permutation chosen so
//   every B component is either a d register or a direct per-lane load; slots
//   with zero weight columns absorb the don't-care copies.  Inner loop:
//   2 loads + 4 WMMAs + adds + 5 tanh -- no selects, no shuffles, no LDS.
//   x is software-pipelined 4 stages deep + global_prefetch_b8 into L2.
// Phase 2: 25-step RNN (18->32 then 32->32), zero-select recurrence.
// Phase 3: 32->3 output projection.
//
// Layouts (16x16x4 f32 WMMA):
//   A:   lane l: row i = l&15,  VGPR0/1 hold K = 2*(l>=16)+{0,1}   (weights)
//   B:   lane l: col j = l&15,  VGPR0/1 hold K(rows) = 2*(l>=16)+{0,1} (data)
//   C/D: lane l: col j = l&15,  VGPR r holds row i = r + 8*(l>=16)
//
// Phase-1 packing: units 0-4 -> D rows 0-4 (lo half), units 5-8 -> rows 8-11
// (hi half).  u-slot table (lo provides slots 4c+{0,1}, hi 4c+{2,3}):
//   k: 0,1 = h0,h1   2,3 = h5,h6    4,5 = h2,h3   6,7  = h7,h8
//      8   = h4      9,10 = ZERO    11  = x4      12..15 = x0,x1,x2,x3
// ---------------------------------------------------------------------------

typedef __attribute__((ext_vector_type(2))) float v2f;
typedef __attribute__((ext_vector_type(8))) float v8f;

#define T_LEN 2048
#define B_N   4096
#define H1N   9
#define H2N   32
#define DIN   5
#define OUTT  25

__device__ __forceinline__ v8f wmma4(v2f a, v2f b, v8f c) {
  return __builtin_amdgcn_wmma_f32_16x16x4_f32(
      /*neg_a=*/false, a, /*neg_b=*/false, b,
      /*c_mod=*/(short)0, c, /*reuse_a=*/false, /*reuse_b=*/false);
}

__device__ __forceinline__ float fast_tanh(float v) {
#if __has_builtin(__builtin_amdgcn_tanhf)
  return __builtin_amdgcn_tanhf(v);   // v_tanh_f32 (gfx1250 trans op)
#else
  return tanhf(v);
#endif
}

// ---------------------------------------------------------------------------
// Phase 1 recurrence step.  d[0..4] holds this half's units (lo: h0-h4,
// hi: h5-h8 in d[0..3], d[4] stays 0).  xp = {x0,x1} (lo) / {x2,x3} (hi)
// loaded per-lane; xw = x4 (zero-weighted on the lo side).
// ---------------------------------------------------------------------------
__device__ __forceinline__ void rnn1_step(
    v2f A0, v2f A1, v2f A2, v2f A3, v8f cbias,
    float d[5], float2 xp, float xw)
{
  v2f B0; B0.x = d[0]; B0.y = d[1];            // rows {0,1}/{2,3}
  v2f B1; B1.x = d[2]; B1.y = d[3];            // rows {4,5}/{6,7}
  v2f B2; B2.x = d[4]; B2.y = xw;              // rows {8,9}/{10,11}
  v2f B3; B3.x = xp.x; B3.y = xp.y;            // rows {12,13}/{14,15}

  v8f cz = {};
  v8f c1 = wmma4(A0, B0, cbias);               // 4 independent accumulators:
  v8f c2 = wmma4(A1, B1, cz);                  // h-dependency depth = 1 WMMA
  v8f c3 = wmma4(A2, B2, cz);
  v8f c4 = wmma4(A3, B3, cz);

#pragma unroll
  for (int i = 0; i < 5; ++i)
    d[i] = fast_tanh((c1[i] + c2[i]) + (c3[i] + c4[i]));
}

template <int DIR>
__device__ __forceinline__ void rnn1_time_loop(
    const float* __restrict__ x, long long xrow, int hi,
    v2f A0, v2f A1, v2f A2, v2f A3, v8f cbias, float d[5])
{
  constexpr int STEP = DIR ? -DIN : DIN;                  // elements per step
  const long long t0 = DIR ? (long long)(T_LEN - 1) * DIN : 0;
  const float* q = x + xrow + t0 + 2 * hi;                // {x0,x1} / {x2,x3}
  const float* r = x + xrow + t0 + 4;                     // x4

#pragma unroll
  for (int i = 0; i < 5; ++i) d[i] = 0.f;                 // h(0) = 0

  // 4-stage x pipeline: stage u holds x(t+u); ~3 steps of load slack
  float2 xp[4]; float xw[4];
#pragma unroll
  for (int u = 0; u < 4; ++u) {
    xp[u] = *(const float2*)&q[u * STEP];
    xw[u] = r[u * STEP];
  }
  q += 4 * STEP; r += 4 * STEP;

#pragma unroll 1
  for (int t = 0; t < T_LEN - 4; t += 4) {
    __builtin_prefetch(r + STEP * 60, 0, 1);              // global_prefetch_b8
#pragma unroll
    for (int u = 0; u < 4; ++u) {
      rnn1_step(A0, A1, A2, A3, cbias, d, xp[u], xw[u]);
      xp[u] = *(const float2*)&q[u * STEP];               // x(t+4+u)
      xw[u] = r[u * STEP];
    }
    q += 4 * STEP; r += 4 * STEP;
  }
#pragma unroll
  for (int u = 0; u < 4; ++u)                             // epilogue
    rnn1_step(A0, A1, A2, A3, cbias, d, xp[u], xw[u]);
}

__global__ __launch_bounds__(32) void rnn1_wmma_kernel(
    const float* __restrict__ x,
    const float* __restrict__ w_ih_f, const float* __restrict__ w_hh_f,
    const float* __restrict__ b_ih_f, const float* __restrict__ b_hh_f,
    const float* __restrict__ w_ih_b, const float* __restrict__ w_hh_b,
    const float* __restrict__ b_ih_b, const float* __restrict__ b_hh_b,
    float* __restrict__ y /* [B_N][18] */)
{
  const int lane = threadIdx.x;        // 0..31
  const int g    = blockIdx.x;         // 0..511
  const int dir  = g >> 8;             // 0 = fwd, 1 = bwd
  const int s0   = (g & 255) * 16;
  const int hi   = lane >> 4;
  const int ii   = lane & 15;          // A row / B,D column (sample)

  const float* wih = dir ? w_ih_b : w_ih_f;
  const float* whh = dir ? w_hh_b : w_hh_f;
  const float* bih = dir ? b_ih_b : b_ih_f;
  const float* bhh = dir ? b_hh_b : b_hh_f;

  // unit hosted by A/D row i: rows 0-4 -> units 0-4, rows 8-11 -> units 5-8
  auto unitOfRow = [](int i) -> int {
    return (i < 5) ? i : ((i >= 8 && i < 12) ? i - 3 : -1);
  };
  // u-slot tables: type 0 = h (Whh column), 1 = x (Wih column), 2 = zero
  const int slotType[16] = {0,0,0,0, 0,0,0,0, 0,2,2,1, 1,1,1,1};
  const int slotIdx [16] = {0,1,5,6, 2,3,7,8, 4,0,0,4, 0,1,2,3};

  const int urow = unitOfRow(ii);
  auto wElem = [&](int k) -> float {
    if (urow < 0) return 0.f;
    const int ty = slotType[k], ix = slotIdx[k];
    if (ty == 0) return whh[urow * H1N + ix];
    if (ty == 1) return wih[urow * DIN + ix];
    return 0.f;
  };
  auto aW = [&](int kb) -> v2f {
    v2f v; const int k0 = kb + 2 * hi;
    v.x = wElem(k0); v.y = wElem(k0 + 1);
    return v;
  };
  const v2f A0 = aW(0), A1 = aW(4), A2 = aW(8), A3 = aW(12);

  // C bias per row i = r + 8*hi
  v8f cbias;
#pragma unroll
  for (int rr = 0; rr < 8; ++rr) {
    const int uu = unitOfRow(rr + 8 * hi);
    cbias[rr] = (uu >= 0) ? (bih[uu] + bhh[uu]) : 0.f;
  }

  const long long xrow = (long long)(s0 + ii) * (T_LEN * DIN);

  float d[5];
  if (dir == 0)
    rnn1_time_loop<0>(x, xrow, hi, A0, A1, A2, A3, cbias, d);
  else
    rnn1_time_loop<1>(x, xrow, hi, A0, A1, A2, A3, cbias, d);

  // final hidden: row r+8hi hosts unit u -> y[s0+j][dir*9 + u]
  const int sj = s0 + ii;
#pragma unroll
  for (int rr = 0; rr < 5; ++rr) {
    const int uu = unitOfRow(rr + 8 * hi);
    if (uu >= 0) y[sj * (2 * H1N) + dir * H1N + uu] = d[rr];
  }
}

// ---------------------------------------------------------------------------
// Phase 2: 25-step RNN, hidden 32; input y only at t=0.  Permuted u ordering
// makes every recurrence B-chunk a plain register pair of d: zero shuffles,
// zero selects.  Slot k=4c+kk maps to unit 16*(c>>2) + 8*(kk>>1) + 2*(c&3)
// + (kk&1); D tile nt row r+8hi hosts unit 16nt + r + 8hi.
// ---------------------------------------------------------------------------
__global__ __launch_bounds__(32) void rnn2_wmma_kernel(
    const float* __restrict__ w_ih2, const float* __restrict__ w_hh2,
    const float* __restrict__ b_ih2, const float* __restrict__ b_hh2,
    const float* __restrict__ y  /* [B_N][18] */,
    float* __restrict__ ys       /* [B_N][25][32] */)
{
  const int lane = threadIdx.x;
  const int hi = lane >> 4, ii = lane & 15;
  const int s0 = blockIdx.x * 16;

  // step-0 weights (identity K ordering over y, K = 18 padded to 20)
  v2f Aih[5][2];
#pragma unroll
  for (int c = 0; c < 5; ++c)
#pragma unroll
    for (int nt = 0; nt < 2; ++nt) {
      const int k0 = 4 * c + 2 * hi, i = nt * 16 + ii;
      Aih[c][nt].x = ((k0    ) < 18) ? w_ih2[i * 18 + k0    ] : 0.f;
      Aih[c][nt].y = ((k0 + 1) < 18) ? w_ih2[i * 18 + k0 + 1] : 0.f;
    }
  // recurrence weights with permuted K: slot 4c+2hi -> unit j0 (and j0+1)
  v2f Ahh[8][2];
#pragma unroll
  for (int c = 0; c < 8; ++c)
#pragma unroll
    for (int nt = 0; nt < 2; ++nt) {
      const int j0 = 16 * (c >> 2) + 8 * hi + 2 * (c & 3);
      const int i = nt * 16 + ii;
      Ahh[c][nt].x = w_hh2[i * 32 + j0    ];
      Ahh[c][nt].y = w_hh2[i * 32 + j0 + 1];
    }

  v8f cb[2];
#pragma unroll
  for (int nt = 0; nt < 2; ++nt)
#pragma unroll
    for (int rr = 0; rr < 8; ++rr) {
      const int i = nt * 16 + rr + 8 * hi;
      cb[nt][rr] = b_ih2[i] + b_hh2[i];
    }

  // step-0 B operands from y; stray (masked) loads stay inside ws
  const int base = (s0 + ii) * 18;
  v2f By[5];
#pragma unroll
  for (int c = 0; c < 5; ++c) {
    const int k0 = 4 * c + 2 * hi;
    const float v0 = y[base + k0];
    const float v1 = y[base + k0 + 1];
    By[c].x = ((k0    ) < 18) ? v0 : 0.f;
    By[c].y = ((k0 + 1) < 18) ? v1 : 0.f;
  }

  v8f d[2];
  v8f cz = {};
#pragma unroll 1
  for (int t = 0; t < OUTT; ++t) {
    if (t == 0) {
#pragma unroll
      for (int nt = 0; nt < 2; ++nt) {
        v8f c1 = wmma4(Aih[0][nt], By[0], cb[nt]);
        v8f c2 = wmma4(Aih[1][nt], By[1], cz);
        c1 = wmma4(Aih[2][nt], By[2], c1);
        c2 = wmma4(Aih[3][nt], By[3], c2);
        c1 = wmma4(Aih[4][nt], By[4], c1);
#pragma unroll
        for (int rr = 0; rr < 8; ++rr) d[nt][rr] = fast_tanh(c1[rr] + c2[rr]);
      }
    } else {
      // B chunks are plain register pairs of d
      v2f Bh[8];
#pragma unroll
      for (int c = 0; c < 8; ++c) {
        Bh[c].x = d[c >> 2][2 * (c & 3)    ];
        Bh[c].y = d[c >> 2][2 * (c & 3) + 1];
      }
      v8f dn[2];
#pragma unroll
      for (int nt = 0; nt < 2; ++nt) {
        v8f c1 = cb[nt];
        v8f c2 = cz;
#pragma unroll
        for (int cc = 0; cc < 8; cc += 2) {
          c1 = wmma4(Ahh[cc    ][nt], Bh[cc    ], c1);
          c2 = wmma4(Ahh[cc + 1][nt], Bh[cc + 1], c2);
        }
#pragma unroll
        for (int rr = 0; rr < 8; ++rr) dn[nt][rr] = fast_tanh(c1[rr] + c2[rr]);
      }
      d[0] = dn[0]; d[1] = dn[1];
    }

    // emit hidden state: D tile nt row r+8hi = unit 16nt+r+8hi of sample j
    const long long ob = ((long long)(s0 + ii) * OUTT + t) * 32;
#pragma unroll
    for (int nt = 0; nt < 2; ++nt)
#pragma unroll
      for (int rr = 0; rr < 8; ++rr)
        ys[ob + nt * 16 + rr + 8 * hi] = d[nt][rr];
  }
}

// ---------------------------------------------------------------------------
// Phase 3: out[b,t,:] = ys[b,t,:] @ w_out.T + b_out   (one lane per (b,t))
// ---------------------------------------------------------------------------
__global__ __launch_bounds__(256) void out_proj_kernel(
    const float* __restrict__ ys, const float* __restrict__ w_out,
    const float* __restrict__ b_out, float* __restrict__ out)
{
  const int id = blockIdx.x * blockDim.x + threadIdx.x;   // (b*25 + t)
  if (id >= B_N * OUTT) return;
  const float* row = ys + (long long)id * 32;
  float a0 = b_out[0], a1 = b_out[1], a2 = b_out[2];
#pragma unroll
  for (int nn = 0; nn < 32; ++nn) {
    const float v = row[nn];
    a0 = fmaf(v, w_out[nn],      a0);
    a1 = fmaf(v, w_out[32 + nn], a1);
    a2 = fmaf(v, w_out[64 + nn], a2);
  }
  out[(long long)id * 3 + 0] = a0;
  out[(long long)id * 3 + 1] = a1;
  out[(long long)id * 3 + 2] = a2;
}

// ---------------------------------------------------------------------------
extern "C" void kernel_launch(void* const* d_in, const int* in_sizes, int n_in,
                              void* d_out, int out_size, void* d_ws,
                              size_t ws_size, hipStream_t stream) {
  const float* x      = (const float*)d_in[0];
  const float* w_ih_f = (const float*)d_in[1];
  const float* w_hh_f = (const float*)d_in[2];
  const float* b_ih_f = (const float*)d_in[3];
  const float* b_hh_f = (const float*)d_in[4];
  const float* w_ih_b = (const float*)d_in[5];
  const float* w_hh_b = (const float*)d_in[6];
  const float* b_ih_b = (const float*)d_in[7];
  const float* b_hh_b = (const float*)d_in[8];
  const float* w_ih2  = (const float*)d_in[9];
  const float* w_hh2  = (const float*)d_in[10];
  const float* b_ih2  = (const float*)d_in[11];
  const float* b_hh2  = (const float*)d_in[12];
  const float* w_out  = (const float*)d_in[13];
  const float* b_out  = (const float*)d_in[14];

  float* y  = (float*)d_ws;                    // [4096][18]
  float* ys = y + (size_t)B_N * (2 * H1N);     // [4096][25][32]

  rnn1_wmma_kernel<<<dim3(512), dim3(32), 0, stream>>>(
      x, w_ih_f, w_hh_f, b_ih_f, b_hh_f,
      w_ih_b, w_hh_b, b_ih_b, b_hh_b, y);

  rnn2_wmma_kernel<<<dim3(256), dim3(32), 0, stream>>>(
      w_ih2, w_hh2, b_ih2, b_hh2, y, ys);

  const int nthreads = B_N * OUTT;             // 102400
  out_proj_kernel<<<dim3((nthreads + 255) / 256), dim3(256), 0, stream>>>(
      ys, w_out, b_out, (float*)d_out);
}